// PixelShift_8744553414923
// MI455X (gfx1250) — compile-verified
//
#include <hip/hip_runtime.h>
#include <stdint.h>

// PixelShift: out[c,d,i,j] = x[c,d,(i - sh0[d]) % H, (j - sh1[d]) % W]
// Pure bandwidth problem: 512 MB traffic -> ~22us floor @ 23.3 TB/s.
// Strategy: per-wave async DMA of the (row-shifted) source row into LDS
// (GLOBAL_LOAD_ASYNC_TO_LDS_B128, ASYNCcnt), then rotated LDS gather +
// aligned non-temporal b128 stores. All global traffic is aligned 128-bit.

#define C_      4
#define D_      16
#define H_      1024
#define W_      1024
#define WAVES_  8                 // waves per block (wave32 -> 256 threads)
#define THREADS (WAVES_ * 32)

typedef float v4f __attribute__((ext_vector_type(4)));

__global__ __launch_bounds__(THREADS)
void PixelShift_kernel(const float* __restrict__ x,
                       const int*   __restrict__ shift,   // [D,2] int32
                       float*       __restrict__ out)
{
    // One 4KB LDS slice per wave; 32 KB total (of 320 KB/WGP).
    __shared__ __align__(16) float smem[WAVES_ * W_];

    const int tid  = threadIdx.x;
    const int wave = tid >> 5;
    const int lane = tid & 31;

    const int cd = blockIdx.y;                 // 0 .. C*D-1  (c = cd/D, d = cd%D)
    const int d  = cd & (D_ - 1);
    const int i  = blockIdx.x * WAVES_ + wave; // output row handled by this wave

    // Per-slice shifts (uniform per block -> scalar loads).
    int sh0 = shift[2 * d + 0];
    int sh1 = shift[2 * d + 1];
    sh0 %= H_; if (sh0 < 0) sh0 += H_;         // normalize to [0,H)
    sh1 %= W_; if (sh1 < 0) sh1 += W_;         // normalize to [0,W)

    int src_row = i - sh0; if (src_row < 0) src_row += H_;

    const float* srcrow = x   + ((size_t)cd * H_ + src_row) * W_;
    float*       dstrow = out + ((size_t)cd * H_ + i)       * W_;

    // ---- Phase 1: async DMA source row -> this wave's LDS slice --------
    // 8 x (32 lanes x 16 B) = 4096 B, contiguous & 16B-aligned in global.
    const uint32_t lds_base = (uint32_t)(uintptr_t)(&smem[wave * W_]);
    #pragma unroll
    for (int k = 0; k < 8; ++k) {
        const uint32_t off  = (uint32_t)(k * 512 + lane * 16);   // byte offset
        const uint32_t ldsa = lds_base + off;                    // LDS byte addr
        const uint64_t gaddr = (uint64_t)(uintptr_t)srcrow + off;
        asm volatile("global_load_async_to_lds_b128 %0, %1, off"
                     :
                     : "v"(ldsa), "v"(gaddr)
                     : "memory");
    }
    // Wave-private consumption: only need this wave's async ops to land.
    asm volatile("s_wait_asynccnt 0" ::: "memory");

    // ---- Phase 2: rotated LDS gather -> aligned b128 NT stores ---------
    const float* wrow = &smem[wave * W_];
    #pragma unroll
    for (int k = 0; k < 8; ++k) {
        const int j0 = k * 128 + lane * 4;     // output column base (16B aligned)
        int c0 = j0     - sh1; if (c0 < 0) c0 += W_;
        int c1 = j0 + 1 - sh1; if (c1 < 0) c1 += W_;
        int c2 = j0 + 2 - sh1; if (c2 < 0) c2 += W_;
        int c3 = j0 + 3 - sh1; if (c3 < 0) c3 += W_;
        v4f v;
        v.x = wrow[c0];
        v.y = wrow[c1];
        v.z = wrow[c2];
        v.w = wrow[c3];
        // Output is write-once/never-read: non-temporal to spare L2 for the
        // 256MB input stream (L2 is 192MB; keep it for reads).
        __builtin_nontemporal_store(v, (v4f*)(dstrow + j0));
    }
}

extern "C" void kernel_launch(void* const* d_in, const int* in_sizes, int n_in,
                              void* d_out, int out_size, void* d_ws, size_t ws_size,
                              hipStream_t stream) {
    const float* x     = (const float*)d_in[0];   // [C,D,H,W] f32
    const int*   shift = (const int*)  d_in[1];   // [D,2] int32
    float*       out   = (float*)      d_out;     // [C,D,H,W] f32

    dim3 grid(H_ / WAVES_, C_ * D_);              // (128, 64)
    PixelShift_kernel<<<grid, THREADS, 0, stream>>>(x, shift, out);
}